// HomeostatL1_3281355014686
// MI455X (gfx1250) — compile-verified
//
#include <hip/hip_runtime.h>
#include <cstdint>

// HomeostatL1: K=16.7M elementwise f32 stream (5 in, 3 out, 536 MB total)
// + 5 global f64-accumulated reductions. Pure bandwidth problem (~23 us at
// 23.3 TB/s). CDNA5 paths used: global_load_async_to_lds_b128 double-buffered
// pipeline with s_wait_asynccnt, ds_load_b128 consumption, non-temporal
// (th:NT) streaming stores since working set >> 192 MB L2.

#define TPB 256
#define GRID_MAX 2048
#define USE_ASYNC_PIPE 1

#define HDT  0.01f
#define HEPS 1e-8f

__device__ __forceinline__ void homeo_elem(
    float uu, float xx, float nn, float sg, float gm,
    float& nx, float& nd, float& dn,
    double& a0, double& a1, double& a2, double& a3, double& a4)
{
    float inv = 1.0f / (sg + HEPS);   // Sigma^-1 diag
    float d   = inv * nn;             // drive
    nd = -gm * d + xx + uu;           // n_dot
    nx = fmaxf(nn + HDT * nd, 0.0f);  // Euler + floor
    dn = inv * nx;                    // d_new
    a0 += (double)(d  * nn);          // n Sigma^-1 n      (F_prev*2)
    a1 += (double)(dn * nx);          // n' Sigma^-1 n'    (F_next*2)
    a2 += (double)(dn * nd);          // dot(d_new, n_dot) (-valence)
    a3 += (double)(dn * dn);          // |d_new|^2         (arousal^2)
    a4 += (double)inv;                // sum Sigma^-1      (-dominance)
}

__device__ __forceinline__ void homeo_vec4(
    float4 U, float4 X, float4 N, float4 S, float4 G, int j,
    float* __restrict__ o_nx, float* __restrict__ o_nd, float* __restrict__ o_dn,
    double& a0, double& a1, double& a2, double& a3, double& a4)
{
    float nx, nd, dn;
    homeo_elem(U.x, X.x, N.x, S.x, G.x, nx, nd, dn, a0, a1, a2, a3, a4);
    __builtin_nontemporal_store(nx, o_nx + j + 0);
    __builtin_nontemporal_store(nd, o_nd + j + 0);
    __builtin_nontemporal_store(dn, o_dn + j + 0);
    homeo_elem(U.y, X.y, N.y, S.y, G.y, nx, nd, dn, a0, a1, a2, a3, a4);
    __builtin_nontemporal_store(nx, o_nx + j + 1);
    __builtin_nontemporal_store(nd, o_nd + j + 1);
    __builtin_nontemporal_store(dn, o_dn + j + 1);
    homeo_elem(U.z, X.z, N.z, S.z, G.z, nx, nd, dn, a0, a1, a2, a3, a4);
    __builtin_nontemporal_store(nx, o_nx + j + 2);
    __builtin_nontemporal_store(nd, o_nd + j + 2);
    __builtin_nontemporal_store(dn, o_dn + j + 2);
    homeo_elem(U.w, X.w, N.w, S.w, G.w, nx, nd, dn, a0, a1, a2, a3, a4);
    __builtin_nontemporal_store(nx, o_nx + j + 3);
    __builtin_nontemporal_store(nd, o_nd + j + 3);
    __builtin_nontemporal_store(dn, o_dn + j + 3);
}

__global__ void __launch_bounds__(TPB) homeo_main(
    const float* __restrict__ u, const float* __restrict__ xi,
    const float* __restrict__ nq, const float* __restrict__ sig,
    const float* __restrict__ gam,
    float* __restrict__ out, double* __restrict__ partials, int K)
{
    const int tid   = (int)threadIdx.x;
    const int K4    = K >> 2;
    const int tiles = K4 / TPB;   // full 256-float4 tiles

    float* __restrict__ o_nx = out + 1;                   // n_next  (addr % 16 == 4)
    float* __restrict__ o_nd = out + 1 + (size_t)K;       // n_dot   (addr % 16 == 4)
    float* __restrict__ o_dn = out + 4 + (size_t)2 * K;   // d_new   (16B aligned)

    double a0 = 0.0, a1 = 0.0, a2 = 0.0, a3 = 0.0, a4 = 0.0;

    __shared__ float4 stage[2][5][TPB];   // 40 KB double-buffered async staging
    __shared__ double red[5][TPB];        // 10 KB block-reduction scratch

#if USE_ASYNC_PIPE
    // Each lane async-copies its own 16B slice of all 5 streams into LDS.
    // 5 async instructions per tile per wave; no cross-lane sharing -> only
    // per-wave s_wait_asynccnt, no workgroup barrier in the hot loop.
#define ISSUE_TILE(tile_, buf_)                                                              \
    do {                                                                                     \
        unsigned goff_ = (((unsigned)(tile_)) * (unsigned)TPB + (unsigned)tid) * 16u;        \
        unsigned l0_ = (unsigned)(uintptr_t)&stage[(buf_)][0][tid];                          \
        unsigned l1_ = (unsigned)(uintptr_t)&stage[(buf_)][1][tid];                          \
        unsigned l2_ = (unsigned)(uintptr_t)&stage[(buf_)][2][tid];                          \
        unsigned l3_ = (unsigned)(uintptr_t)&stage[(buf_)][3][tid];                          \
        unsigned l4_ = (unsigned)(uintptr_t)&stage[(buf_)][4][tid];                          \
        asm volatile("global_load_async_to_lds_b128 %0, %1, %2"                              \
                     :: "v"(l0_), "v"(goff_), "s"(u)   : "memory");                          \
        asm volatile("global_load_async_to_lds_b128 %0, %1, %2"                              \
                     :: "v"(l1_), "v"(goff_), "s"(xi)  : "memory");                          \
        asm volatile("global_load_async_to_lds_b128 %0, %1, %2"                              \
                     :: "v"(l2_), "v"(goff_), "s"(nq)  : "memory");                          \
        asm volatile("global_load_async_to_lds_b128 %0, %1, %2"                              \
                     :: "v"(l3_), "v"(goff_), "s"(sig) : "memory");                          \
        asm volatile("global_load_async_to_lds_b128 %0, %1, %2"                              \
                     :: "v"(l4_), "v"(goff_), "s"(gam) : "memory");                          \
    } while (0)

    {
        int t = (int)blockIdx.x;
        if (t < tiles) {
            int buf = 0;
            ISSUE_TILE(t, 0);
            for (;;) {
                int tn = t + (int)gridDim.x;
                if (tn < tiles) {
                    ISSUE_TILE(tn, buf ^ 1);
                    // 10 outstanding; wait until next tile's 5 remain
                    asm volatile("s_wait_asynccnt 5" ::: "memory");
                } else {
                    asm volatile("s_wait_asynccnt 0" ::: "memory");
                }
                float4 U = stage[buf][0][tid];
                float4 X = stage[buf][1][tid];
                float4 N = stage[buf][2][tid];
                float4 S = stage[buf][3][tid];
                float4 G = stage[buf][4][tid];
                homeo_vec4(U, X, N, S, G, (t * TPB + tid) * 4,
                           o_nx, o_nd, o_dn, a0, a1, a2, a3, a4);
                if (tn >= tiles) break;
                t = tn;
                buf ^= 1;
            }
        }
    }
#undef ISSUE_TILE
#else
    {
        const float4* u4 = (const float4*)u;
        const float4* x4 = (const float4*)xi;
        const float4* n4 = (const float4*)nq;
        const float4* s4 = (const float4*)sig;
        const float4* g4 = (const float4*)gam;
        for (int t = (int)blockIdx.x; t < tiles; t += (int)gridDim.x) {
            int i = t * TPB + tid;
            homeo_vec4(u4[i], x4[i], n4[i], s4[i], g4[i], i * 4,
                       o_nx, o_nd, o_dn, a0, a1, a2, a3, a4);
        }
    }
#endif

    // Tail (K not a multiple of 1024): handled by block 0 with direct loads.
    if (blockIdx.x == 0) {
        const float4* u4 = (const float4*)u;
        const float4* x4 = (const float4*)xi;
        const float4* n4 = (const float4*)nq;
        const float4* s4 = (const float4*)sig;
        const float4* g4 = (const float4*)gam;
        for (int i = tiles * TPB + tid; i < K4; i += TPB)
            homeo_vec4(u4[i], x4[i], n4[i], s4[i], g4[i], i * 4,
                       o_nx, o_nd, o_dn, a0, a1, a2, a3, a4);
        for (int i = (K4 << 2) + tid; i < K; i += TPB) {
            float nx, nd, dn;
            homeo_elem(u[i], xi[i], nq[i], sig[i], gam[i], nx, nd, dn,
                       a0, a1, a2, a3, a4);
            o_nx[i] = nx; o_nd[i] = nd; o_dn[i] = dn;
        }
    }

    // Deterministic block reduction (f64) -> per-block partials in d_ws.
    __syncthreads();
    red[0][tid] = a0; red[1][tid] = a1; red[2][tid] = a2;
    red[3][tid] = a3; red[4][tid] = a4;
    __syncthreads();
    for (int s = TPB / 2; s > 0; s >>= 1) {
        if (tid < s) {
#pragma unroll
            for (int c = 0; c < 5; ++c) red[c][tid] += red[c][tid + s];
        }
        __syncthreads();
    }
    if (tid == 0) {
#pragma unroll
        for (int c = 0; c < 5; ++c)
            partials[(size_t)blockIdx.x * 5 + c] = red[c][0];
    }
}

__global__ void __launch_bounds__(TPB) homeo_finalize(
    const double* __restrict__ partials, int nblocks,
    float* __restrict__ out, int K)
{
    __shared__ double red[5][TPB];
    const int tid = (int)threadIdx.x;
    double a[5] = {0.0, 0.0, 0.0, 0.0, 0.0};
    for (int b = tid; b < nblocks; b += TPB) {
#pragma unroll
        for (int c = 0; c < 5; ++c) a[c] += partials[(size_t)b * 5 + c];
    }
#pragma unroll
    for (int c = 0; c < 5; ++c) red[c][tid] = a[c];
    __syncthreads();
    for (int s = TPB / 2; s > 0; s >>= 1) {
        if (tid < s) {
#pragma unroll
            for (int c = 0; c < 5; ++c) red[c][tid] += red[c][tid + s];
        }
        __syncthreads();
    }
    if (tid == 0) {
        double s_nn  = red[0][0];   // n  Sigma^-1 n
        double s_nn2 = red[1][0];   // n' Sigma^-1 n'
        double s_vd  = red[2][0];   // dot(d_new, n_dot)
        double s_dd  = red[3][0];   // |d_new|^2
        double s_inv = red[4][0];   // sum Sigma^-1
        out[0]                    = (float)(0.5 * (s_nn - s_nn2)); // reward
        out[1 + (size_t)2 * K]    = (float)(-s_vd);                // valence
        out[2 + (size_t)2 * K]    = (float)sqrt(s_dd);             // arousal
        out[3 + (size_t)2 * K]    = (float)(-s_inv);               // dominance
        out[4 + (size_t)3 * K]    = (float)(0.5 * s_nn2);          // F_next
    }
}

extern "C" void kernel_launch(void* const* d_in, const int* in_sizes, int n_in,
                              void* d_out, int out_size, void* d_ws, size_t ws_size,
                              hipStream_t stream)
{
    const float* u   = (const float*)d_in[0];
    const float* xi  = (const float*)d_in[1];
    const float* nq  = (const float*)d_in[2];
    const float* sig = (const float*)d_in[3];
    const float* gam = (const float*)d_in[4];
    float* out = (float*)d_out;
    const int K  = in_sizes[0];
    const int K4 = K >> 2;
    const int tiles = K4 / TPB;

    int grid = (tiles < GRID_MAX) ? (tiles > 0 ? tiles : 1) : GRID_MAX;
    size_t need = (size_t)grid * 5 * sizeof(double);
    while (grid > 1 && need > ws_size) {
        grid >>= 1;
        need = (size_t)grid * 5 * sizeof(double);
    }
    double* partials = (double*)d_ws;

    homeo_main<<<grid, TPB, 0, stream>>>(u, xi, nq, sig, gam, out, partials, K);
    homeo_finalize<<<1, TPB, 0, stream>>>(partials, grid, out, K);
}